// GNN_55293408968886
// MI455X (gfx1250) — compile-verified
//
#include <hip/hip_runtime.h>
#include <hip/hip_bf16.h>

typedef __attribute__((ext_vector_type(2))) float v2f;
typedef __attribute__((ext_vector_type(8))) float v8f;

#define LAT 128   // latent width (N of every GEMM)
#define NT  8     // 8 column tiles of 16 -> 128 outputs per wave

__device__ __forceinline__ v8f wmma4(v2f a, v2f b, v8f c) {
  return __builtin_amdgcn_wmma_f32_16x16x4_f32(
      /*neg_a=*/false, a, /*neg_b=*/false, b,
      /*c_mod=*/(short)0, c, /*reuse_a=*/false, /*reuse_b=*/false);
}

// One k-step (K advances by 4) of a 32(M)x128(N) fp32 WMMA tile:
// two 16-row A fragments share every B fragment -> 16 WMMAs per 18 loads.
__device__ __forceinline__ void wmma_step2(v8f* acc0, v8f* acc1,
                                           const float* __restrict__ a0p,
                                           const float* __restrict__ a1p,
                                           const float* __restrict__ wrow,
                                           int kb, int col) {
  v2f a0; a0.x = a0p[kb]; a0.y = a0p[kb + 1];
  v2f a1; a1.x = a1p[kb]; a1.y = a1p[kb + 1];
#pragma unroll
  for (int t = 0; t < NT; ++t) {
    v2f b;
    b.x = wrow[kb * LAT + t * 16 + col];
    b.y = wrow[(kb + 1) * LAT + t * 16 + col];
    acc0[t] = wmma4(a0, b, acc0[t]);
    acc1[t] = wmma4(a1, b, acc1[t]);
  }
}

// ---------------- zero init (float4) ----------------
__global__ __launch_bounds__(256) void k_zero4(float4* __restrict__ p, size_t n4) {
  size_t i = (size_t)blockIdx.x * blockDim.x + threadIdx.x;
  size_t stride = (size_t)gridDim.x * blockDim.x;
  float4 z = {0.f, 0.f, 0.f, 0.f};
  for (; i < n4; i += stride) p[i] = z;
}

// ---------------- fused weight precompute ----------------
// Wf[16,128] = W_ee[16,128] @ W_eu[0:128, 0:128]
// bf[128]    = b_eu + b_ee @ W_eu[0:128, 0:128]
__global__ __launch_bounds__(256) void k_fuse(const float* __restrict__ W_ee,
                                              const float* __restrict__ b_ee,
                                              const float* __restrict__ W_eu,
                                              const float* __restrict__ b_eu,
                                              float* __restrict__ Wf,
                                              float* __restrict__ bf) {
  int idx = blockIdx.x * 256 + threadIdx.x;
  if (idx < 16 * LAT) {
    int i = idx >> 7, j = idx & 127;
    float s = 0.f;
#pragma unroll 4
    for (int l = 0; l < LAT; ++l) s += W_ee[i * LAT + l] * W_eu[l * LAT + j];
    Wf[idx] = s;
  } else if (idx < 16 * LAT + LAT) {
    int j = idx - 16 * LAT;
    float s = b_eu[j];
#pragma unroll 4
    for (int l = 0; l < LAT; ++l) s += b_ee[l] * W_eu[l * LAT + j];
    bf[j] = s;
  }
}

// ---------------- node encoder: C[M,128] = A[M,K] @ W[K,128] + bias ----------------
// 32 rows per wave (two fragments), wave-uniform tail handling.
__global__ __launch_bounds__(256) void k_node_encoder(const float* __restrict__ A,
                                                      const float* __restrict__ W,
                                                      const float* __restrict__ bias,
                                                      float* __restrict__ C,
                                                      int M, int K) {
  int lane = threadIdx.x & 31;
  int gw = blockIdx.x * 8 + (threadIdx.x >> 5);
  long m0 = (long)gw * 32;
  if (m0 >= M) return;                       // wave-uniform
  bool hasB = (m0 + 16) < M;                 // wave-uniform (M % 16 == 0)
  int row = lane & 15;
  int col = lane & 15;
  int kb = (lane >> 4) << 1;

  v8f acc0[NT] = {}, acc1[NT] = {};
  const float* aA = A + (size_t)(m0 + row) * K;
  const float* aB = A + (size_t)(m0 + (hasB ? 16 : 0) + row) * K;  // clamp: safe loads
#pragma unroll 2
  for (int k = 0; k < K; k += 4)
    wmma_step2(acc0, acc1, aA + k, aB + k, W + (size_t)k * LAT, kb, col);

  int h8 = (lane >> 4) * 8;
#pragma unroll
  for (int t = 0; t < NT; ++t) {
    float bv = bias[t * 16 + col];
#pragma unroll
    for (int r = 0; r < 8; ++r)
      C[(size_t)(m0 + r + h8) * LAT + t * 16 + col] = acc0[t][r] + bv;
    if (hasB) {
#pragma unroll
      for (int r = 0; r < 8; ++r)
        C[(size_t)(m0 + 16 + r + h8) * LAT + t * 16 + col] = acc1[t][r] + bv;
    }
  }
}

// ---------------- edge update (fused encoder seg0) + atomic scatter ----------------
// 32 edges per wave; B fragments shared across both edge fragments.
__global__ __launch_bounds__(256) void k_edge_update(const float* __restrict__ ef,   // [E,16]
                                                     const float* __restrict__ nl,   // [N,128]
                                                     const float* __restrict__ g,    // [16]
                                                     const int* __restrict__ snd,
                                                     const int* __restrict__ rcv,
                                                     const float* __restrict__ Wf,   // [16,128]
                                                     const float* __restrict__ bf,   // [128]
                                                     const float* __restrict__ Weu,  // [400,128]
                                                     float* __restrict__ sagg,
                                                     float* __restrict__ ragg,
                                                     int E) {
  int lane = threadIdx.x & 31;
  int gw = blockIdx.x * 8 + (threadIdx.x >> 5);
  long e0 = (long)gw * 32;
  if (e0 >= E) return;                       // wave-uniform (E % 32 == 0)
  int row = lane & 15;
  int col = lane & 15;
  int kb = (lane >> 4) << 1;
  long eA = e0 + row;
  long eB = e0 + 16 + row;
  int saA = snd[eA], raA = rcv[eA];
  int saB = snd[eB], raB = rcv[eB];

  v8f acc0[NT] = {}, acc1[NT] = {};
  // seg0: raw edge features through fused weights (K=16)
  {
    const float* pA = ef + (size_t)eA * 16;
    const float* pB = ef + (size_t)eB * 16;
#pragma unroll
    for (int k = 0; k < 16; k += 4)
      wmma_step2(acc0, acc1, pA + k, pB + k, Wf + (size_t)k * LAT, kb, col);
  }
  // seg1: sender node latent (K=128), W_eu rows 128..255
  {
    const float* pA = nl + (size_t)saA * LAT;
    const float* pB = nl + (size_t)saB * LAT;
#pragma unroll 2
    for (int k = 0; k < 128; k += 4)
      wmma_step2(acc0, acc1, pA + k, pB + k, Weu + (size_t)(128 + k) * LAT, kb, col);
  }
  // seg2: receiver node latent (K=128), W_eu rows 256..383
  {
    const float* pA = nl + (size_t)raA * LAT;
    const float* pB = nl + (size_t)raB * LAT;
#pragma unroll 2
    for (int k = 0; k < 128; k += 4)
      wmma_step2(acc0, acc1, pA + k, pB + k, Weu + (size_t)(256 + k) * LAT, kb, col);
  }
  // seg3: globals (K=16, identical rows -> duplicate loads CSE away)
#pragma unroll
  for (int k = 0; k < 16; k += 4)
    wmma_step2(acc0, acc1, g + k, g + k, Weu + (size_t)(384 + k) * LAT, kb, col);

  // fused segment-sum: scatter new_edges into sent/recv aggregates
  float bv[NT];
#pragma unroll
  for (int t = 0; t < NT; ++t) bv[t] = bf[t * 16 + col];
  int h8 = (lane >> 4) * 8;
#pragma unroll
  for (int f = 0; f < 2; ++f) {
    const v8f* acc = f ? acc1 : acc0;
#pragma unroll
    for (int r = 0; r < 8; ++r) {
      long e = e0 + f * 16 + r + h8;
      size_t so = (size_t)snd[e] * LAT;
      size_t ro = (size_t)rcv[e] * LAT;
#pragma unroll
      for (int t = 0; t < NT; ++t) {
        float v = acc[t][r] + bv[t];
        atomicAdd(sagg + so + t * 16 + col, v);
        atomicAdd(ragg + ro + t * 16 + col, v);
      }
    }
  }
}

// ---------------- node update: concat(nodes, sent, recv, g) @ W_nu + b_nu ----------------
__global__ __launch_bounds__(256) void k_node_update(const float* __restrict__ nl,
                                                     const float* __restrict__ sagg,
                                                     const float* __restrict__ ragg,
                                                     const float* __restrict__ g,
                                                     const float* __restrict__ W,  // [400,128]
                                                     const float* __restrict__ bias,
                                                     float* __restrict__ out,
                                                     int M) {
  int lane = threadIdx.x & 31;
  int gw = blockIdx.x * 8 + (threadIdx.x >> 5);
  long m0 = (long)gw * 32;
  if (m0 >= M) return;                       // wave-uniform
  bool hasB = (m0 + 16) < M;                 // wave-uniform
  int row = lane & 15;
  int col = lane & 15;
  int kb = (lane >> 4) << 1;
  size_t rowA = (size_t)(m0 + row) * LAT;
  size_t rowB = (size_t)(m0 + (hasB ? 16 : 0) + row) * LAT;  // clamp: safe loads

  v8f acc0[NT] = {}, acc1[NT] = {};
#pragma unroll 2
  for (int k = 0; k < 128; k += 4)
    wmma_step2(acc0, acc1, nl + rowA + k, nl + rowB + k, W + (size_t)k * LAT, kb, col);
#pragma unroll 2
  for (int k = 0; k < 128; k += 4)
    wmma_step2(acc0, acc1, sagg + rowA + k, sagg + rowB + k, W + (size_t)(128 + k) * LAT, kb, col);
#pragma unroll 2
  for (int k = 0; k < 128; k += 4)
    wmma_step2(acc0, acc1, ragg + rowA + k, ragg + rowB + k, W + (size_t)(256 + k) * LAT, kb, col);
#pragma unroll
  for (int k = 0; k < 16; k += 4)
    wmma_step2(acc0, acc1, g + k, g + k, W + (size_t)(384 + k) * LAT, kb, col);

  int h8 = (lane >> 4) * 8;
#pragma unroll
  for (int t = 0; t < NT; ++t) {
    float bv = bias[t * 16 + col];
#pragma unroll
    for (int r = 0; r < 8; ++r)
      out[(size_t)(m0 + r + h8) * LAT + t * 16 + col] = acc0[t][r] + bv;
    if (hasB) {
#pragma unroll
      for (int r = 0; r < 8; ++r)
        out[(size_t)(m0 + 16 + r + h8) * LAT + t * 16 + col] = acc1[t][r] + bv;
    }
  }
}

// ---------------- output head: [M,128] @ [128,5] + b ----------------
__global__ __launch_bounds__(256) void k_out_head(const float* __restrict__ nn,
                                                  const float* __restrict__ Wo,
                                                  const float* __restrict__ bo,
                                                  float* __restrict__ out,
                                                  int M) {
  __shared__ float w[LAT * 5];
  for (int i = threadIdx.x; i < LAT * 5; i += 256) w[i] = Wo[i];
  __syncthreads();
  int n = blockIdx.x * 256 + threadIdx.x;
  if (n >= M) return;
  float a0 = bo[0], a1 = bo[1], a2 = bo[2], a3 = bo[3], a4 = bo[4];
  const float* x = nn + (size_t)n * LAT;
#pragma unroll 4
  for (int k = 0; k < LAT; ++k) {
    float xv = x[k];
    a0 += xv * w[k * 5 + 0];
    a1 += xv * w[k * 5 + 1];
    a2 += xv * w[k * 5 + 2];
    a3 += xv * w[k * 5 + 3];
    a4 += xv * w[k * 5 + 4];
  }
  float* o = out + (size_t)n * 5;
  o[0] = a0; o[1] = a1; o[2] = a2; o[3] = a3; o[4] = a4;
}

extern "C" void kernel_launch(void* const* d_in, const int* in_sizes, int n_in,
                              void* d_out, int out_size, void* d_ws, size_t ws_size,
                              hipStream_t stream) {
  const float* node_feats = (const float*)d_in[0];
  const float* edge_feats = (const float*)d_in[1];
  const float* globals_   = (const float*)d_in[2];
  const int*   senders    = (const int*)d_in[3];
  const int*   receivers  = (const int*)d_in[4];
  const float* W_ne  = (const float*)d_in[5];
  const float* b_ne  = (const float*)d_in[6];
  const float* W_ee  = (const float*)d_in[7];
  const float* b_ee  = (const float*)d_in[8];
  const float* W_eu  = (const float*)d_in[9];
  const float* b_eu  = (const float*)d_in[10];
  const float* W_nu  = (const float*)d_in[11];
  const float* b_nu  = (const float*)d_in[12];
  const float* W_out = (const float*)d_in[13];
  const float* b_out = (const float*)d_in[14];
  float* out = (float*)d_out;

  const int N = 50000, E = 600000;
  const size_t NODE_BYTES = (size_t)N * LAT * sizeof(float);  // 25.6 MB

  char* ws = (char*)d_ws;
  float* nodes_lat = (float*)(ws + 0 * NODE_BYTES);
  float* sagg      = (float*)(ws + 1 * NODE_BYTES);
  float* ragg      = (float*)(ws + 2 * NODE_BYTES);  // contiguous after sagg
  float* new_nodes = (float*)(ws + 3 * NODE_BYTES);
  float* Wf        = (float*)(ws + 4 * NODE_BYTES);
  float* bf        = (float*)(ws + 4 * NODE_BYTES + 16 * LAT * sizeof(float));

  // 1) zero both aggregate buffers (contiguous) — must happen every call
  k_zero4<<<2048, 256, 0, stream>>>((float4*)sagg, (size_t)2 * N * LAT / 4);

  // 2) fuse edge encoder into edge-update segment-0 weights
  k_fuse<<<9, 256, 0, stream>>>(W_ee, b_ee, W_eu, b_eu, Wf, bf);

  // 3) node encoder: ceil(50000/32) = 1563 wave tiles, 8 waves per block
  k_node_encoder<<<(1563 + 7) / 8, 256, 0, stream>>>(node_feats, W_ne, b_ne, nodes_lat, N, 128);

  // 4) edge update + fused segment-sum scatter: 600000/32 = 18750 wave tiles
  k_edge_update<<<(18750 + 7) / 8, 256, 0, stream>>>(edge_feats, nodes_lat, globals_,
                                                     senders, receivers, Wf, bf, W_eu,
                                                     sagg, ragg, E);

  // 5) node update
  k_node_update<<<(1563 + 7) / 8, 256, 0, stream>>>(nodes_lat, sagg, ragg, globals_,
                                                    W_nu, b_nu, new_nodes, N);

  // 6) output head
  k_out_head<<<(N + 255) / 256, 256, 0, stream>>>(new_nodes, W_out, b_out, out, N);
}